// Date_Conditioning_Block_38182259261863
// MI455X (gfx1250) — compile-verified
//
#include <hip/hip_runtime.h>

typedef __attribute__((ext_vector_type(2))) float v2f;
typedef __attribute__((ext_vector_type(8))) float v8f;

#define N_CHANNELS 256
#define HIDDEN 32
#define BATCH 16384

// gelu(x) = 0.5*x*(1+erf(x/sqrt(2))), erf via Abramowitz-Stegun 7.1.26
__device__ __forceinline__ float gelu_erf(float h) {
  const float kInvSqrt2 = 0.70710678118654752f;
  float z  = h * kInvSqrt2;
  float az = __builtin_fabsf(z);
  float t  = __builtin_amdgcn_rcpf(__builtin_fmaf(0.3275911f, az, 1.0f));
  float p  = __builtin_fmaf(1.061405429f, t, -1.453152027f);
  p = __builtin_fmaf(p, t, 1.421413741f);
  p = __builtin_fmaf(p, t, -0.284496736f);
  p = __builtin_fmaf(p, t, 0.254829592f);
  p *= t;
  float e       = __expf(-z * z);
  float erf_abs = __builtin_fmaf(-p, e, 1.0f);
  float erfz    = (z < 0.0f) ? -erf_abs : erf_abs;
  return 0.5f * h * (1.0f + erfz);
}

__global__ __launch_bounds__(256) void per_channel_mlp_kernel(
    const float* __restrict__ x,   // [BATCH, 2]
    const float* __restrict__ W0,  // [C, 32, 2]
    const float* __restrict__ b0,  // [C, 32]
    const float* __restrict__ W1,  // [C, 2, 32]
    const float* __restrict__ b1,  // [C, 2]
    float* __restrict__ out)       // [BATCH, C, 2]
{
  const int c    = blockIdx.x;                       // channel
  const int lane = threadIdx.x & 31;
  const int waveId = blockIdx.y * (blockDim.x >> 5) + (threadIdx.x >> 5); // 0..63
  const bool lo  = lane < 16;
  const int  jr  = lane & 15;                        // row within 16-row tile / batch col
  const int  hi8 = (lane >> 4) << 3;                 // 0 or 8: accumulator M offset

  // ---- hoisted per-channel weights (loop-invariant) ----
  // Layer-0 A-matrix (16x4 f32, K=0..1 real, K=2..3 zero-pad), two 16-row tiles.
  // A layout: VGPR0 = K0 (lanes 0-15) | K2 (lanes 16-31); VGPR1 = K1 | K3.
  const float2 w0p0 = *(const float2*)(W0 + (size_t)c * 64 + (size_t)jr * 2);
  const float2 w0p1 = *(const float2*)(W0 + (size_t)c * 64 + (size_t)(jr + 16) * 2);
  v2f a0, a1;
  a0.x = lo ? w0p0.x : 0.0f;  a0.y = lo ? w0p0.y : 0.0f;
  a1.x = lo ? w0p1.x : 0.0f;  a1.y = lo ? w0p1.y : 0.0f;

  // Layer-0 bias as the WMMA C operand: C[m=j, n] = b0[c,j] for all n.
  // C/D layout: VGPR v, lanes 0-15 -> M=v ; lanes 16-31 -> M=v+8.
  v8f c0, c1;
#pragma unroll
  for (int v = 0; v < 8; ++v) {
    c0[v] = b0[c * HIDDEN + hi8 + v];
    c1[v] = b0[c * HIDDEN + 16 + hi8 + v];
  }

  // Layer-1 weights for this lane's 16 hidden indices j = {t*16 + hi8 + v}.
  float w1r[2][2][8];
#pragma unroll
  for (int o = 0; o < 2; ++o)
#pragma unroll
    for (int tt = 0; tt < 2; ++tt)
#pragma unroll
      for (int v = 0; v < 8; ++v)
        w1r[o][tt][v] = W1[c * 64 + o * HIDDEN + tt * 16 + hi8 + v];

  const float2 bb = *(const float2*)(b1 + (size_t)c * 2);

  float2* __restrict__ out2 = (float2*)out;          // out[b][c][0:2] as float2

  for (int t = 0; t < 16; ++t) {
    const int bbase = waveId * 256 + t * 16;

    // B-matrix (4x16 f32): B[k, n=batch col]; K=0..1 = x, K=2..3 zero-pad.
    // B layout mirrors A: VGPR0 = K0 | K2 ; VGPR1 = K1 | K3.
    const float2 xv = *(const float2*)(x + (size_t)(bbase + jr) * 2);
    v2f bm;
    bm.x = lo ? xv.x : 0.0f;
    bm.y = lo ? xv.y : 0.0f;

    // h^T tiles: D[j, b'] = sum_k W0[c,j,k] * x[b,k] + b0[c,j]
    v8f h0 = __builtin_amdgcn_wmma_f32_16x16x4_f32(false, a0, false, bm,
                                                   (short)0, c0, false, false);
    v8f h1 = __builtin_amdgcn_wmma_f32_16x16x4_f32(false, a1, false, bm,
                                                   (short)0, c1, false, false);

    // GELU + layer-1 (K=32 reduction, 16 j's per lane) in VALU.
    float s0 = 0.0f, s1 = 0.0f;
#pragma unroll
    for (int v = 0; v < 8; ++v) {
      float g0 = gelu_erf(h0[v]);
      float g1 = gelu_erf(h1[v]);
      s0 = __builtin_fmaf(g0, w1r[0][0][v], s0);
      s0 = __builtin_fmaf(g1, w1r[0][1][v], s0);
      s1 = __builtin_fmaf(g0, w1r[1][0][v], s1);
      s1 = __builtin_fmaf(g1, w1r[1][1][v], s1);
    }
    // combine the two 8-j halves (lane l <-> l^16 hold same batch col)
    s0 += __shfl_xor(s0, 16, 32);
    s1 += __shfl_xor(s1, 16, 32);

    if (lane < 16) {
      float2 r;
      r.x = s0 + bb.x;
      r.y = s1 + bb.y;
      out2[(size_t)(bbase + lane) * N_CHANNELS + c] = r;
    }
  }
}

extern "C" void kernel_launch(void* const* d_in, const int* in_sizes, int n_in,
                              void* d_out, int out_size, void* d_ws, size_t ws_size,
                              hipStream_t stream) {
  const float* x  = (const float*)d_in[0];
  const float* W0 = (const float*)d_in[1];
  const float* b0 = (const float*)d_in[2];
  const float* W1 = (const float*)d_in[3];
  const float* b1 = (const float*)d_in[4];
  float* out = (float*)d_out;

  // 256 channels x 8 blocks; 8 waves/block -> 64 waves per channel,
  // each wave loops over 16 tiles of 16 batch rows = 16384 rows total.
  dim3 grid(N_CHANNELS, 8);
  per_channel_mlp_kernel<<<grid, 256, 0, stream>>>(x, W0, b0, W1, b1, out);
}